// DSGAT_GraphBranch_25744033973188
// MI455X (gfx1250) — compile-verified
//
#include <hip/hip_runtime.h>
#include <hip/hip_bf16.h>

// ---------------------------------------------------------------------------
// Constants from the reference
// ---------------------------------------------------------------------------
#define C_DIM   256   // IN_C == D1 == H*HID
#define H_HEADS 4
#define HID_DIM 64
#define NEG_SLOPE 0.2f
#define LP_ALPHA  0.5f
#define LDS_STRIDE (C_DIM + 4)   // pad to avoid 64-bank conflicts

typedef float v2f __attribute__((ext_vector_type(2)));
typedef float v8f __attribute__((ext_vector_type(8)));
typedef int   v4i_b __attribute__((vector_size(16)));   // async-LDS builtin payload

#if defined(__gfx1250__) && __has_builtin(__builtin_amdgcn_global_load_async_to_lds_b128)
#define USE_ASYNC_LDS 1
#else
#define USE_ASYNC_LDS 0
#endif

// ---------------------------------------------------------------------------
// Helpers
// ---------------------------------------------------------------------------
__device__ __forceinline__ void atomicMaxFloat(float* addr, float val) {
    unsigned int* ia = (unsigned int*)addr;
    unsigned int old = __hip_atomic_load(ia, __ATOMIC_RELAXED, __HIP_MEMORY_SCOPE_AGENT);
    while (__uint_as_float(old) < val) {
        unsigned int assumed = old;
        old = atomicCAS(ia, assumed, __float_as_uint(val));
        if (old == assumed) break;
    }
}

__global__ void fill_k(float* __restrict__ p, float v, long long n) {
    long long i = (long long)blockIdx.x * blockDim.x + threadIdx.x;
    if (i < n) p[i] = v;
}

// ---------------------------------------------------------------------------
// fp32 GEMM via V_WMMA_F32_16X16X4_F32:  out[M,256] = A[M,256] @ W[256,256]
// One block = 16-row M tile; 8 waves x (two 16-col tiles) = 256 cols.
// A tile staged in LDS (16 x 256, padded). On CDNA5 the staging uses the
// async direct-to-LDS path (GLOBAL_LOAD_ASYNC_TO_LDS_B128, ASYNCcnt);
// W fragments stream from the L2-resident 256KB weight matrix with a
// global_prefetch one K-step ahead.
// ---------------------------------------------------------------------------
__global__ __launch_bounds__(256) void gemm_f32_wmma(
        const float* __restrict__ A, const float* __restrict__ W,
        float* __restrict__ out) {
    __shared__ float As[16 * LDS_STRIDE];
    const int tid  = threadIdx.x;
    const int lane = tid & 31;
    const int wave = tid >> 5;
    const int mBase = blockIdx.x * 16;

    // stage 16x256 fp32 A tile into LDS: 1024 x 16B chunks, 256 threads
#if USE_ASYNC_LDS
    for (int i = tid; i < 16 * (C_DIM / 4); i += 256) {
        int row  = i >> 6;       // 64 x 16B chunks per row
        int col4 = i & 63;
        const float* g = A + (size_t)(mBase + row) * C_DIM + col4 * 4;
        float* l = &As[row * LDS_STRIDE + col4 * 4];
        __builtin_amdgcn_global_load_async_to_lds_b128(
            (__attribute__((address_space(1))) v4i_b*)(v4i_b*)g,
            (__attribute__((address_space(3))) v4i_b*)(v4i_b*)l, 0, 0);
    }
    asm volatile("s_wait_asynccnt 0" ::: "memory");
    __syncthreads();
#else
    for (int i = tid; i < 16 * (C_DIM / 4); i += 256) {
        int row  = i >> 6;
        int col4 = i & 63;
        float4 v = ((const float4*)(A + (size_t)(mBase + row) * C_DIM))[col4];
        float* d = &As[row * LDS_STRIDE + col4 * 4];
        d[0] = v.x; d[1] = v.y; d[2] = v.z; d[3] = v.w;
    }
    __syncthreads();
#endif

    // WMMA fragment coordinates (wave32, 16x16x4 f32):
    //   A: lane -> m = lane&15, k = kstep + (lane>>4)*2 (+0/+1 in .x/.y)
    //   B: lane -> n = lane&15, same k mapping
    //   C/D: vgpr i -> m = i + (lane>>4)*8, n = lane&15
    const int m    = lane & 15;
    const int koff = (lane >> 4) * 2;
    const int nCol = wave * 32 + (lane & 15);

    v8f c0 = {}; v8f c1 = {};
    for (int k = 0; k < C_DIM; k += 4) {
        v2f a;
        a.x = As[m * LDS_STRIDE + k + koff];
        a.y = As[m * LDS_STRIDE + k + koff + 1];
        const float* Wr0 = W + (size_t)(k + koff) * C_DIM;
        const float* Wr1 = W + (size_t)(k + koff + 1) * C_DIM;
        if (k + 4 < C_DIM)   // pull next K rows of W toward the WGP
            __builtin_prefetch(W + (size_t)(k + 4 + koff) * C_DIM + nCol, 0, 3);
        v2f b0, b1;
        b0.x = Wr0[nCol];      b0.y = Wr1[nCol];
        b1.x = Wr0[nCol + 16]; b1.y = Wr1[nCol + 16];
        c0 = __builtin_amdgcn_wmma_f32_16x16x4_f32(false, a, false, b0,
                                                   (short)0, c0, false, false);
        c1 = __builtin_amdgcn_wmma_f32_16x16x4_f32(false, a, false, b1,
                                                   (short)0, c1, false, false);
    }

    const int mHi = (lane >> 4) * 8;
    for (int i = 0; i < 8; ++i) {
        size_t row = (size_t)(mBase + mHi + i) * C_DIM;
        out[row + wave * 32 + (lane & 15)]      = c0[i];
        out[row + wave * 32 + 16 + (lane & 15)] = c1[i];
    }
}

// ---------------------------------------------------------------------------
// Per-node attention logits: es[n,h] = sum_c h[n,h,c]*a_src[h,c] (ditto ed)
// ---------------------------------------------------------------------------
__global__ void attn_logits(const float* __restrict__ hpre,
                            const float* __restrict__ a_s,
                            const float* __restrict__ a_d,
                            float* __restrict__ es, float* __restrict__ ed,
                            int n) {
    int i = blockIdx.x * blockDim.x + threadIdx.x;
    if (i >= n) return;
    const float* row = hpre + (size_t)i * C_DIM;
    #pragma unroll
    for (int h = 0; h < H_HEADS; ++h) {
        float s = 0.f, d = 0.f;
        #pragma unroll 8
        for (int c = 0; c < HID_DIM; ++c) {
            float v = row[h * HID_DIM + c];
            s += v * a_s[h * HID_DIM + c];
            d += v * a_d[h * HID_DIM + c];
        }
        es[i * H_HEADS + h] = s;
        ed[i * H_HEADS + h] = d;
    }
}

// ---------------------------------------------------------------------------
// GAT edge passes (edges 0..E-1 from edge_index, E..E+N-1 are self-loops)
// ---------------------------------------------------------------------------
__device__ __forceinline__ void edge_nodes(long long i, const int* src,
                                           const int* dst, int E, int* s, int* d) {
    if (i < E) { *s = src[i]; *d = dst[i]; }
    else       { *s = *d = (int)(i - E); }
}

__global__ void edge_attn_max(const int* __restrict__ src, const int* __restrict__ dst,
                              const float* __restrict__ es, const float* __restrict__ ed,
                              float* __restrict__ mbuf, int E, int N) {
    long long i = (long long)blockIdx.x * blockDim.x + threadIdx.x;
    if (i >= (long long)E + N) return;
    int s, d; edge_nodes(i, src, dst, E, &s, &d);
    #pragma unroll
    for (int h = 0; h < H_HEADS; ++h) {
        float e = es[s * H_HEADS + h] + ed[d * H_HEADS + h];
        e = e > 0.f ? e : NEG_SLOPE * e;
        atomicMaxFloat(&mbuf[d * H_HEADS + h], e);
    }
}

__global__ void edge_expsum(const int* __restrict__ src, const int* __restrict__ dst,
                            const float* __restrict__ es, const float* __restrict__ ed,
                            const float* __restrict__ mbuf, float* __restrict__ sbuf,
                            int E, int N) {
    long long i = (long long)blockIdx.x * blockDim.x + threadIdx.x;
    if (i >= (long long)E + N) return;
    int s, d; edge_nodes(i, src, dst, E, &s, &d);
    #pragma unroll
    for (int h = 0; h < H_HEADS; ++h) {
        float e = es[s * H_HEADS + h] + ed[d * H_HEADS + h];
        e = e > 0.f ? e : NEG_SLOPE * e;
        atomicAdd(&sbuf[d * H_HEADS + h], expf(e - mbuf[d * H_HEADS + h]));
    }
}

// one wave per edge: 256-wide weighted scatter-add (8 float atomics / lane)
__global__ __launch_bounds__(256) void edge_scatter(
        const int* __restrict__ src, const int* __restrict__ dst,
        const float* __restrict__ es, const float* __restrict__ ed,
        const float* __restrict__ mbuf, const float* __restrict__ sbuf,
        const float* __restrict__ hpre, float* __restrict__ acc, int E, int N) {
    const int lane = threadIdx.x & 31;
    long long i = (long long)blockIdx.x * 8 + (threadIdx.x >> 5);
    if (i >= (long long)E + N) return;
    int s, d; edge_nodes(i, src, dst, E, &s, &d);
    float alpha[H_HEADS];
    #pragma unroll
    for (int h = 0; h < H_HEADS; ++h) {
        float e = es[s * H_HEADS + h] + ed[d * H_HEADS + h];
        e = e > 0.f ? e : NEG_SLOPE * e;
        float p = expf(e - mbuf[d * H_HEADS + h]);
        alpha[h] = p / (sbuf[d * H_HEADS + h] + 1e-16f);
    }
    const float* hs = hpre + (size_t)s * C_DIM;
    float* ao = acc + (size_t)d * C_DIM;
    #pragma unroll
    for (int it = 0; it < 8; ++it) {
        int c = lane + it * 32;
        atomicAdd(&ao[c], hs[c] * alpha[c >> 6]);
    }
}

__global__ void bias_elu(const float* __restrict__ acc, const float* __restrict__ b,
                         float* __restrict__ out, long long n) {
    long long i = (long long)blockIdx.x * blockDim.x + threadIdx.x;
    if (i >= n) return;
    float v = acc[i] + b[i & (C_DIM - 1)];
    out[i] = v > 0.f ? v : (expf(v) - 1.0f);
}

// ---------------------------------------------------------------------------
// Label propagation
// ---------------------------------------------------------------------------
__global__ void deg_count(const int* __restrict__ dst, float* __restrict__ deg, int E) {
    int i = blockIdx.x * blockDim.x + threadIdx.x;
    if (i < E) atomicAdd(&deg[dst[i]], 1.0f);
}
__global__ void deg_to_dis(float* __restrict__ deg, int n) {
    int i = blockIdx.x * blockDim.x + threadIdx.x;
    if (i < n) {
        float d = deg[i];
        deg[i] = d > 0.f ? rsqrtf(fmaxf(d, 1e-12f)) : 0.f;
    }
}
__global__ void scale_half(const float* __restrict__ y, float* __restrict__ res,
                           long long n) {
    long long i = (long long)blockIdx.x * blockDim.x + threadIdx.x;
    if (i < n) res[i] = (1.0f - LP_ALPHA) * y[i];
}
__global__ __launch_bounds__(256) void lp_scatter(
        const int* __restrict__ src, const int* __restrict__ dst,
        const float* __restrict__ dis, const float* __restrict__ y,
        float* __restrict__ tmp, int E) {
    const int lane = threadIdx.x & 31;
    long long i = (long long)blockIdx.x * 8 + (threadIdx.x >> 5);
    if (i >= E) return;
    int s = src[i], d = dst[i];
    float w = dis[s] * dis[d];
    const float* ys = y + (size_t)s * C_DIM;
    float* td = tmp + (size_t)d * C_DIM;
    #pragma unroll
    for (int it = 0; it < 8; ++it) {
        int c = lane + it * 32;
        atomicAdd(&td[c], ys[c] * w);
    }
}
__global__ void lp_update(const float* __restrict__ tmp, const float* __restrict__ res,
                          float* __restrict__ y, long long n) {
    long long i = (long long)blockIdx.x * blockDim.x + threadIdx.x;
    if (i >= n) return;
    float v = LP_ALPHA * tmp[i] + res[i];
    y[i] = fminf(fmaxf(v, 0.f), 1.f);
}

// ---------------------------------------------------------------------------
// Pooling + MLP head
// ---------------------------------------------------------------------------
__global__ __launch_bounds__(256) void pool_sum(
        const float* __restrict__ x, const float* __restrict__ h1,
        const float* __restrict__ h2, const int* __restrict__ batch,
        float* __restrict__ pooled, float* __restrict__ counts, int N) {
    const int lane = threadIdx.x & 31;
    long long node = (long long)blockIdx.x * 8 + (threadIdx.x >> 5);
    if (node >= N) return;
    int g = batch[node];
    if (lane == 0) atomicAdd(&counts[g], 1.0f);
    float* pg = pooled + (size_t)g * 768;
    #pragma unroll
    for (int it = 0; it < 8; ++it) {
        int c = lane + it * 32;
        atomicAdd(&pg[c],       x [(size_t)node * C_DIM + c]);
        atomicAdd(&pg[256 + c], h1[(size_t)node * C_DIM + c]);
        atomicAdd(&pg[512 + c], h2[(size_t)node * C_DIM + c]);
    }
}
__global__ void emb_div(float* __restrict__ pooled, const float* __restrict__ counts,
                        int G) {
    int i = blockIdx.x * blockDim.x + threadIdx.x;
    if (i >= G * 768) return;
    pooled[i] /= fmaxf(counts[i / 768], 1.0f);
}
__global__ void mlp1_k(const float* __restrict__ emb, const float* __restrict__ w,
                       const float* __restrict__ b, float* __restrict__ hidden) {
    int j = threadIdx.x;          // 0..255
    int g = blockIdx.x;
    float a = b[j];
    const float* eg = emb + (size_t)g * 768;
    for (int k = 0; k < 768; ++k) a += eg[k] * w[(size_t)k * 256 + j];
    hidden[(size_t)g * 256 + j] = fmaxf(a, 0.f);
}
__global__ void mlp2_k(const float* __restrict__ hidden, const float* __restrict__ w,
                       const float* __restrict__ b, float* __restrict__ out) {
    int j = threadIdx.x;          // 0..127
    int g = blockIdx.x;
    float a = b[j];
    const float* hg = hidden + (size_t)g * 256;
    for (int k = 0; k < 256; ++k) a += hg[k] * w[(size_t)k * 128 + j];
    out[(size_t)g * 128 + j] = a;
}

// ---------------------------------------------------------------------------
// Launch
// ---------------------------------------------------------------------------
extern "C" void kernel_launch(void* const* d_in, const int* in_sizes, int n_in,
                              void* d_out, int out_size, void* d_ws, size_t ws_size,
                              hipStream_t stream) {
    const float* x   = (const float*)d_in[0];
    const int*   ei  = (const int*)  d_in[1];
    const int*   bat = (const int*)  d_in[2];
    const float* W1  = (const float*)d_in[3];
    const float* as1 = (const float*)d_in[4];
    const float* ad1 = (const float*)d_in[5];
    const float* b1  = (const float*)d_in[6];
    const float* W2  = (const float*)d_in[7];
    const float* as2 = (const float*)d_in[8];
    const float* ad2 = (const float*)d_in[9];
    const float* b2  = (const float*)d_in[10];
    const float* mw1 = (const float*)d_in[11];
    const float* mb1 = (const float*)d_in[12];
    const float* mw2 = (const float*)d_in[13];
    const float* mb2 = (const float*)d_in[14];
    float* out = (float*)d_out;

    const int N = in_sizes[0] / C_DIM;
    const int E = in_sizes[1] / 2;
    const int G = out_size / 128;
    const int* src = ei;
    const int* dst = ei + E;
    const int TOT = E + N;

    float* ws = (float*)d_ws;
    size_t off = 0;
    auto alloc = [&](size_t n) { float* p = ws + off; off += n; return p; };
    float* hpre   = alloc((size_t)N * C_DIM);   // GEMM out; doubles as LP res
    float* acc    = alloc((size_t)N * C_DIM);   // GAT accumulator; doubles as LP tmp
    float* h1     = alloc((size_t)N * C_DIM);
    float* h2     = alloc((size_t)N * C_DIM);
    float* es     = alloc((size_t)N * H_HEADS);
    float* ed     = alloc((size_t)N * H_HEADS);
    float* mbuf   = alloc((size_t)N * H_HEADS);
    float* sbuf   = alloc((size_t)N * H_HEADS);
    float* dis    = alloc(N);
    float* pooled = alloc((size_t)G * 768);
    float* counts = alloc(G);
    float* hidden = alloc((size_t)G * 256);

    auto cdiv = [](long long a, long long b) { return (int)((a + b - 1) / b); };
    const long long NC = (long long)N * C_DIM;
    const long long NH = (long long)N * H_HEADS;

    // symmetric gcn_norm degree (shared by both label-prop stages)
    fill_k<<<cdiv(N, 256), 256, 0, stream>>>(dis, 0.f, N);
    deg_count<<<cdiv(E, 256), 256, 0, stream>>>(dst, dis, E);
    deg_to_dis<<<cdiv(N, 256), 256, 0, stream>>>(dis, N);

    auto gat = [&](const float* xin, const float* W, const float* a_s,
                   const float* a_d, const float* bb, float* yout) {
        gemm_f32_wmma<<<N / 16, 256, 0, stream>>>(xin, W, hpre);
        attn_logits<<<cdiv(N, 128), 128, 0, stream>>>(hpre, a_s, a_d, es, ed, N);
        fill_k<<<cdiv(NH, 256), 256, 0, stream>>>(mbuf, -3.0e38f, NH);
        edge_attn_max<<<cdiv(TOT, 256), 256, 0, stream>>>(src, dst, es, ed, mbuf, E, N);
        fill_k<<<cdiv(NH, 256), 256, 0, stream>>>(sbuf, 0.f, NH);
        edge_expsum<<<cdiv(TOT, 256), 256, 0, stream>>>(src, dst, es, ed, mbuf, sbuf, E, N);
        fill_k<<<cdiv(NC, 256), 256, 0, stream>>>(acc, 0.f, NC);
        edge_scatter<<<cdiv(TOT, 8), 256, 0, stream>>>(src, dst, es, ed, mbuf, sbuf,
                                                       hpre, acc, E, N);
        bias_elu<<<cdiv(NC, 256), 256, 0, stream>>>(acc, bb, yout, NC);
    };
    auto lprop = [&](float* y) {
        scale_half<<<cdiv(NC, 256), 256, 0, stream>>>(y, hpre, NC);  // res = (1-a)*y
        for (int it = 0; it < 2; ++it) {
            fill_k<<<cdiv(NC, 256), 256, 0, stream>>>(acc, 0.f, NC);
            lp_scatter<<<cdiv(E, 8), 256, 0, stream>>>(src, dst, dis, y, acc, E);
            lp_update<<<cdiv(NC, 256), 256, 0, stream>>>(acc, hpre, y, NC);
        }
    };

    gat(x, W1, as1, ad1, b1, h1);
    lprop(h1);
    gat(h1, W2, as2, ad2, b2, h2);
    lprop(h2);

    fill_k<<<cdiv((long long)G * 768, 256), 256, 0, stream>>>(pooled, 0.f, (long long)G * 768);
    fill_k<<<1, 256, 0, stream>>>(counts, 0.f, G);
    pool_sum<<<cdiv(N, 8), 256, 0, stream>>>(x, h1, h2, bat, pooled, counts, N);
    emb_div<<<cdiv((long long)G * 768, 256), 256, 0, stream>>>(pooled, counts, G);
    mlp1_k<<<G, 256, 0, stream>>>(pooled, mw1, mb1, hidden);
    mlp2_k<<<G, 128, 0, stream>>>(hidden, mw2, mb2, out);
}